// TeamEncoder_21208548508128
// MI455X (gfx1250) — compile-verified
//
#include <hip/hip_runtime.h>
#include <hip/hip_bf16.h>
#include <math.h>

// ---------------- problem constants ----------------
#define NN   4096   // nodes
#define FF   128    // in-features
#define HH   32     // hidden
#define CC   205    // clusters
#define CP   208    // padded clusters (13*16)
#define DD   269    // 2H + C
#define DP   272    // padded D (17*16)
#define LL   512    // teams
#define TT   12     // team size
#define RR   4      // replaced members

#define LDK  36     // LDS row pitch (floats): 16B-aligned rows, conflict-free b64 reads
#define KCH  32     // K-chunk staged per iteration

typedef __attribute__((ext_vector_type(2))) float v2f;
typedef __attribute__((ext_vector_type(8))) float v8f;

#if defined(__AMDGCN__) && __has_builtin(__builtin_amdgcn_wmma_f32_16x16x4_f32)
#define WMMA_F32_OK 1
#else
#define WMMA_F32_OK 0
#endif

#if defined(__AMDGCN__) && __has_builtin(__builtin_amdgcn_global_load_async_to_lds_b128) && __has_builtin(__builtin_amdgcn_s_wait_asynccnt)
#define ASYNC_OK 1
typedef int v4i_vs __attribute__((vector_size(4 * sizeof(int))));
#else
#define ASYNC_OK 0
#endif

// ---------------- wave / block reduction helpers ----------------
__device__ __forceinline__ float wave_sum32(float v) {
    for (int o = 16; o > 0; o >>= 1) v += __shfl_xor(v, o, 32);
    return v;
}
// sum across the 16 lanes that share the same half (bit4 preserved)
__device__ __forceinline__ float half_sum16(float v) {
    for (int o = 8; o > 0; o >>= 1) v += __shfl_xor(v, o, 32);
    return v;
}
__device__ __forceinline__ float bsum(float v, float* sm) {
    int t = threadIdx.x;
    sm[t] = v; __syncthreads();
    for (int s = blockDim.x >> 1; s > 0; s >>= 1) {
        if (t < s) sm[t] += sm[t + s];
        __syncthreads();
    }
    v = sm[0]; __syncthreads();
    return v;
}
__device__ __forceinline__ float bmax(float v, float* sm) {
    int t = threadIdx.x;
    sm[t] = v; __syncthreads();
    for (int s = blockDim.x >> 1; s > 0; s >>= 1) {
        if (t < s) sm[t] = fmaxf(sm[t], sm[t + s]);
        __syncthreads();
    }
    v = sm[0]; __syncthreads();
    return v;
}

// ---------------- direct-global 16x16 WMMA tile accumulators ----------------
// acc fragment mapping: lane l (lane&15) owns column bn+l; vgpr i owns row
// bm + (lane>=16 ? 8 : 0) + i.

// C += A[bm:bm+16, :K] * B[:K, bn:bn+16]   (both row-major)
__device__ __forceinline__ void mma_acc_nn(v8f& acc,
        const float* __restrict__ A, int lda,
        const float* __restrict__ B, int ldb,
        int bm, int bn, int K, int lane) {
    int half = lane >> 4, l = lane & 15;
#if WMMA_F32_OK
    const float* ar = A + (bm + l) * lda + 2 * half;   // K = 2*half, 2*half+1
    const float* br = B + bn + l;
    for (int k = 0; k < K; k += 4) {
        v2f a, b;
        a[0] = ar[k];
        a[1] = ar[k + 1];
        b[0] = br[(k + 2 * half) * ldb];
        b[1] = br[(k + 2 * half + 1) * ldb];
        acc = __builtin_amdgcn_wmma_f32_16x16x4_f32(
                false, a, false, b, (short)0, acc, false, false);
    }
#else
    int m0 = bm + half * 8, cn = bn + l;
    for (int i = 0; i < 8; ++i) {
        float s = acc[i];
        const float* ar = A + (m0 + i) * lda;
        for (int k = 0; k < K; ++k) s = fmaf(ar[k], B[k * ldb + cn], s);
        acc[i] = s;
    }
#endif
}

// ---------------- LDS-staged 128x64 block-tile machinery (V * V^T) ----------
// Stage nrows rows of V[row0.., kc0..kc0+kcw) into LDS tile [nrows][LDK].
// nrows*(kcw/4) is always a multiple of 256 here -> no divergence.
__device__ __forceinline__ void stage_tile(const float* __restrict__ V, int ldv,
                                           int row0, int nrows, int kc0, int kcw,
                                           float* sbase) {
    int spr = kcw >> 2;                       // 16B segments per row
    int nseg = nrows * spr;
    for (int s = threadIdx.x; s < nseg; s += 256) {
        int r = s / spr, q = (s - r * spr) * 4;
        const float* g = V + (row0 + r) * ldv + kc0 + q;
        float* lp = sbase + r * LDK + q;
#if ASYNC_OK
        __builtin_amdgcn_global_load_async_to_lds_b128(
                (v4i_vs*)g, (v4i_vs*)lp, 0, 0);
#else
        lp[0] = g[0]; lp[1] = g[1]; lp[2] = g[2]; lp[3] = g[3];
#endif
    }
}
__device__ __forceinline__ void stage_wait() {
#if ASYNC_OK
    __builtin_amdgcn_s_wait_asynccnt(0);
#endif
    __syncthreads();
}

// One K-chunk of WMMAs from LDS: wave's 16 rows (SAw = SA + wave*16*LDK)
// against 4 column tiles of 16 in SB. acc[c] covers columns c*16..c*16+15.
__device__ __forceinline__ void mma_chunk(v8f acc[4], const float* SAw,
                                          const float* SB, int kcw, int lane) {
    int half = lane >> 4, l = lane & 15;
#if WMMA_F32_OK
    const float* ar = SAw + l * LDK + 2 * half;
    for (int kk = 0; kk < kcw; kk += 4) {
        v2f a;
        a[0] = ar[kk]; a[1] = ar[kk + 1];
        for (int c = 0; c < 4; ++c) {
            const float* br = SB + (c * 16 + l) * LDK + 2 * half;
            v2f b;
            b[0] = br[kk]; b[1] = br[kk + 1];
            acc[c] = __builtin_amdgcn_wmma_f32_16x16x4_f32(
                    false, a, false, b, (short)0, acc[c], false, false);
        }
    }
#else
    for (int c = 0; c < 4; ++c)
        for (int i = 0; i < 8; ++i) {
            float s = acc[c][i];
            const float* ar = SAw + (half * 8 + i) * LDK;
            const float* br = SB + (c * 16 + l) * LDK;
            for (int k = 0; k < kcw; ++k) s = fmaf(ar[k], br[k], s);
            acc[c][i] = s;
        }
#endif
}

// acc[4] += V[bm0:bm0+128 (this wave's 16 rows), :K] * V[bn0:bn0+64, :K]^T
__device__ __forceinline__ void vvt_tile(v8f acc[4],
                                         const float* __restrict__ V, int ldv, int K,
                                         int bm0, int bn0,
                                         float* SA, float* SB) {
    int w = threadIdx.x >> 5, lane = threadIdx.x & 31;
    for (int kc0 = 0; kc0 < K; kc0 += KCH) {
        int kcw = min(KCH, K - kc0);
        __syncthreads();                          // previous chunk fully consumed
        stage_tile(V, ldv, bm0, 128, kc0, kcw, SA);
        stage_tile(V, ldv, bn0, 64,  kc0, kcw, SB);
        stage_wait();
        mma_chunk(acc, SA + w * 16 * LDK, SB, kcw, lane);
    }
}

// ---------------- kernels ----------------
__global__ void k_zero(float* p, int n) {
    int i = blockIdx.x * blockDim.x + threadIdx.x;
    if (i < n) p[i] = 0.f;
}

__global__ void k_pad_wl1(const float* __restrict__ Wl1, float* __restrict__ WP) {
    int i = blockIdx.x * blockDim.x + threadIdx.x;
    if (i >= DD * CC) return;
    int r = i / CC, c = i % CC;
    WP[r * CP + c] = Wl1[i];
}

// C[M,Nc] = A[M,K] * B[K,Nc], one wave per 16x16 tile (tower + logits GEMMs)
__global__ void k_gemm_nn(const float* __restrict__ A, int lda,
                          const float* __restrict__ B, int ldb,
                          float* __restrict__ C, int ldc,
                          int M, int Nc, int K) {
    int wave = (blockIdx.x * blockDim.x + threadIdx.x) >> 5;
    int lane = threadIdx.x & 31;
    int tilesN = Nc >> 4;
    int total = (M >> 4) * tilesN;
    if (wave >= total) return;
    int tm = wave / tilesN, tn = wave % tilesN;
    v8f acc = {0.f, 0.f, 0.f, 0.f, 0.f, 0.f, 0.f, 0.f};
    mma_acc_nn(acc, A, lda, B, ldb, tm * 16, tn * 16, K, lane);
    int half = lane >> 4, l = lane & 15;
    int m0 = tm * 16 + half * 8, cn = tn * 16 + l;
    for (int i = 0; i < 8; ++i) C[(m0 + i) * ldc + cn] = acc[i];
}

// PRE[n,c] = relu( Y[n,:]@Wr[:,c] + X[n,:]@Wroot[:,c] + bias[c] )
__global__ void k_conv_combine(const float* __restrict__ Y, int ldy,
                               const float* __restrict__ X, int ldx,
                               const float* __restrict__ Wr,
                               const float* __restrict__ Wroot,
                               const float* __restrict__ bias,
                               float* __restrict__ OUT, int ldo,
                               int Cc, int K) {
    int idx = blockIdx.x * blockDim.x + threadIdx.x;
    if (idx >= NN * Cc) return;
    int n = idx / Cc, c = idx % Cc;
    float s = bias[c];
    const float* yr = Y + n * ldy;
    const float* xr = X + n * ldx;
    for (int k = 0; k < K; ++k) s = fmaf(yr[k], Wr[k * Cc + c], s);
    for (int k = 0; k < K; ++k) s = fmaf(xr[k], Wroot[k * Cc + c], s);
    OUT[n * ldo + c] = fmaxf(s, 0.f);
}

// per-channel mean / inv-std over N rows (biased variance, eps=1e-5)
__global__ void k_bn_stats(const float* __restrict__ PRE, int ldp,
                           float* __restrict__ mean, float* __restrict__ istd) {
    __shared__ float sm[256];
    int c = blockIdx.x, tid = threadIdx.x;
    float s = 0.f, sq = 0.f;
    for (int n = tid; n < NN; n += blockDim.x) {
        float v = PRE[n * ldp + c];
        s += v; sq += v * v;
    }
    s = bsum(s, sm);
    sq = bsum(sq, sm);
    if (tid == 0) {
        float m = s / (float)NN;
        float var = sq / (float)NN - m * m;
        mean[c] = m;
        istd[c] = rsqrtf(var + 1e-5f);
    }
}

__global__ void k_bn_apply(const float* __restrict__ PRE, int ldp,
                           const float* __restrict__ mean, const float* __restrict__ istd,
                           const float* __restrict__ gamma, const float* __restrict__ beta,
                           float* __restrict__ XC, int xoff, int Cc) {
    int idx = blockIdx.x * blockDim.x + threadIdx.x;
    if (idx >= NN * Cc) return;
    int n = idx / Cc, c = idx % Cc;
    float v = (PRE[n * ldp + c] - mean[c]) * istd[c] * gamma[c] + beta[c];
    XC[n * DP + xoff + c] = v;
}

// per row: bias+relu+softmax -> s1 (to d_out & padded ws), re-softmax entropy,
// row-normalized s1 -> SNP
__global__ void k_softmax_row(float* __restrict__ S1P,
                              const float* __restrict__ bl1,
                              float* __restrict__ outS1,
                              float* __restrict__ SNP,
                              float* __restrict__ accE1) {
    __shared__ float sm[256];
    int n = blockIdx.x, tid = threadIdx.x;
    bool act = tid < CC;
    float t = 0.f;
    if (act) t = fmaxf(S1P[n * CP + tid] + bl1[tid], 0.f);
    float m = bmax(act ? t : -1e30f, sm);
    float e = act ? __expf(t - m) : 0.f;
    float ssum = bsum(e, sm);
    float s1 = act ? e / ssum : 0.f;
    if (tid < CP) S1P[n * CP + tid] = s1;         // pad cols -> 0
    if (act) outS1[n * CC + tid] = s1;
    // entropy of softmax(s1)
    float m2 = bmax(act ? s1 : -1e30f, sm);
    float e2 = act ? __expf(s1 - m2) : 0.f;
    float s2sum = bsum(e2, sm);
    float s2 = act ? e2 / s2sum : 0.f;
    float ent = act ? (-s2 * __logf(s2 + 1e-15f)) : 0.f;
    ent = bsum(ent, sm);
    if (tid == 0) atomicAdd(accE1, ent);
    // row-normalized s1
    float ss = bsum(s1 * s1, sm);
    float inv = 1.f / fmaxf(sqrtf(ss), 1e-8f);
    if (tid < CP) SNP[n * CP + tid] = s1 * inv;
}

__global__ void k_rownorm_x(const float* __restrict__ X, float* __restrict__ XNo) {
    __shared__ float sm[128];
    int n = blockIdx.x, tid = threadIdx.x;
    float v = X[n * FF + tid];
    float ss = bsum(v * v, sm);
    float inv = 1.f / fmaxf(sqrtf(ss), 1e-8f);
    XNo[n * FF + tid] = v * inv;
}

// sum over all (i,j) of (s1 s1^T - adj)^2, LDS-staged 128x64 block tiles
__global__ void __launch_bounds__(256) k_normloss(const float* __restrict__ S1P,
                                                  const float* __restrict__ adj,
                                                  float* __restrict__ acc) {
    __shared__ float SA[128 * LDK];
    __shared__ float SB[64 * LDK];
    int tm = blockIdx.x / (NN / 64), tn = blockIdx.x % (NN / 64);
    int bm0 = tm * 128, bn0 = tn * 64;
    v8f a4[4];
    for (int c = 0; c < 4; ++c) a4[c] = (v8f){0.f, 0.f, 0.f, 0.f, 0.f, 0.f, 0.f, 0.f};
    vvt_tile(a4, S1P, CP, CP, bm0, bn0, SA, SB);
    int w = threadIdx.x >> 5, lane = threadIdx.x & 31;
    int half = lane >> 4, l = lane & 15;
    int m0 = bm0 + w * 16 + half * 8;
    float loc = 0.f;
    for (int c = 0; c < 4; ++c) {
        int cn = bn0 + c * 16 + l;
        for (int i = 0; i < 8; ++i) {
            float d = a4[c][i] - adj[(m0 + i) * NN + cn];
            loc = fmaf(d, d, loc);
        }
    }
    loc = wave_sum32(loc);
    if (lane == 0) atomicAdd(acc, loc);
}

// row sums of squares of V V^T, LDS-staged 128x64 block tiles
__global__ void __launch_bounds__(256) k_rowss(const float* __restrict__ V, int ldv, int K,
                                               float* __restrict__ rs) {
    __shared__ float SA[128 * LDK];
    __shared__ float SB[64 * LDK];
    int tm = blockIdx.x / (NN / 64), tn = blockIdx.x % (NN / 64);
    int bm0 = tm * 128, bn0 = tn * 64;
    v8f a4[4];
    for (int c = 0; c < 4; ++c) a4[c] = (v8f){0.f, 0.f, 0.f, 0.f, 0.f, 0.f, 0.f, 0.f};
    vvt_tile(a4, V, ldv, K, bm0, bn0, SA, SB);
    int w = threadIdx.x >> 5, lane = threadIdx.x & 31;
    int half = lane >> 4, l = lane & 15;
    int m0 = bm0 + w * 16 + half * 8;
    for (int i = 0; i < 8; ++i) {
        float v = 0.f;
        for (int c = 0; c < 4; ++c) v = fmaf(a4[c][i], a4[c][i], v);
        v = half_sum16(v);                        // reduce over 16 columns
        if (l == 0) atomicAdd(rs + m0 + i, v);
    }
}

__global__ void k_invnorm(float* __restrict__ rsx, float* __restrict__ rss) {
    int i = blockIdx.x * blockDim.x + threadIdx.x;
    if (i < NN) rsx[i] = 1.f / fmaxf(sqrtf(rsx[i]), 1e-8f);
    else if (i < 2 * NN) rss[i - NN] = 1.f / fmaxf(sqrtf(rss[i - NN]), 1e-8f);
}

// sum_ij (xn xn^T)[i,j]*(sn sn^T)[i,j]*invx[i]*invs[i], both recomputed via LDS tiles
__global__ void __launch_bounds__(256) k_simdot(const float* __restrict__ XNo,
                                                const float* __restrict__ SNP,
                                                const float* __restrict__ invx,
                                                const float* __restrict__ invs,
                                                float* __restrict__ acc) {
    __shared__ float SA[128 * LDK];
    __shared__ float SB[64 * LDK];
    int tm = blockIdx.x / (NN / 64), tn = blockIdx.x % (NN / 64);
    int bm0 = tm * 128, bn0 = tn * 64;
    v8f ax[4], as[4];
    for (int c = 0; c < 4; ++c) {
        ax[c] = (v8f){0.f, 0.f, 0.f, 0.f, 0.f, 0.f, 0.f, 0.f};
        as[c] = (v8f){0.f, 0.f, 0.f, 0.f, 0.f, 0.f, 0.f, 0.f};
    }
    vvt_tile(ax, XNo, FF, FF, bm0, bn0, SA, SB);
    vvt_tile(as, SNP, CP, CP, bm0, bn0, SA, SB);
    int w = threadIdx.x >> 5, lane = threadIdx.x & 31;
    int half = lane >> 4;
    int m0 = bm0 + w * 16 + half * 8;
    float loc = 0.f;
    for (int c = 0; c < 4; ++c)
        for (int i = 0; i < 8; ++i)
            loc = fmaf(ax[c][i] * as[c][i], invx[m0 + i] * invs[m0 + i], loc);
    loc = wave_sum32(loc);
    if (lane == 0) atomicAdd(acc, loc);
}

__global__ void k_embloss(const float* __restrict__ XC,
                          const int* __restrict__ teams,
                          const int* __restrict__ reps,
                          float* __restrict__ acc) {
    __shared__ float sm[256];
    int t = blockIdx.x, tid = threadIdx.x;
    float loc = 0.f;
    for (int d = tid; d < DD; d += blockDim.x) {
        float rp = 0.f;
        for (int j = 0; j < RR; ++j) rp += XC[reps[t * RR + j] * DP + d];
        rp *= (1.f / RR);
        float rm = 0.f;
        for (int j = RR; j < TT; ++j) rm += XC[teams[t * TT + j] * DP + d];
        rm *= (1.f / (TT - RR));
        loc += fabsf(rp - rm);
    }
    loc = bsum(loc, sm);
    if (tid == 0) atomicAdd(acc, loc);
}

__global__ void k_write_emb(const float* __restrict__ XC, float* __restrict__ out) {
    int i = blockIdx.x * blockDim.x + threadIdx.x;
    if (i >= NN * DD) return;
    int n = i / DD, d = i % DD;
    out[i] = XC[n * DP + d];
}

__global__ void k_finalize(const float* __restrict__ accs, float* __restrict__ out5) {
    float nrm = sqrtf(accs[0]);
    float e1 = accs[1] / (float)NN;
    float sim = -accs[2] / (float)NN;
    float embl = accs[3] / (float)LL;
    out5[0] = nrm;
    out5[1] = e1;
    out5[2] = sim;
    out5[3] = 100.f * nrm + 10.f * e1 + 100.f * sim + embl;
    out5[4] = embl;
}

// ---------------- host launcher ----------------
extern "C" void kernel_launch(void* const* d_in, const int* in_sizes, int n_in,
                              void* d_out, int out_size, void* d_ws, size_t ws_size,
                              hipStream_t stream) {
    const float* x      = (const float*)d_in[0];
    const float* adj    = (const float*)d_in[1];
    const int*   teams  = (const int*)d_in[2];
    const int*   reps   = (const int*)d_in[3];
    const float* W1r    = (const float*)d_in[4];
    const float* b1     = (const float*)d_in[5];
    const float* W1root = (const float*)d_in[6];
    const float* g1     = (const float*)d_in[7];
    const float* be1    = (const float*)d_in[8];
    const float* W2r    = (const float*)d_in[9];
    const float* b2     = (const float*)d_in[10];
    const float* W2root = (const float*)d_in[11];
    const float* g2     = (const float*)d_in[12];
    const float* be2    = (const float*)d_in[13];
    const float* W3r    = (const float*)d_in[14];
    const float* b3     = (const float*)d_in[15];
    const float* W3root = (const float*)d_in[16];
    const float* g3     = (const float*)d_in[17];
    const float* be3    = (const float*)d_in[18];
    const float* Wl1    = (const float*)d_in[19];
    const float* bl1    = (const float*)d_in[20];

    float* out = (float*)d_out;
    float* ws  = (float*)d_ws;

    // workspace layout (floats)
    float* Y    = ws;                        // N*F
    float* PRE  = Y    + NN * FF;            // N*CP
    float* S1P  = PRE  + NN * CP;            // N*CP
    float* SNP  = S1P  + NN * CP;            // N*CP
    float* XNo  = SNP  + NN * CP;            // N*F
    float* MEAN = XNo  + NN * FF;            // 256
    float* ISTD = MEAN + 256;                // 256
    float* XC   = ISTD + 256;                // N*DP  (zeroed)
    float* WL1P = XC   + NN * DP;            // DP*CP (zeroed)
    float* RSX  = WL1P + DP * CP;            // N     (zeroed)
    float* RSS  = RSX  + NN;                 // N     (zeroed)
    float* ACCS = RSS  + NN;                 // 8     (zeroed)
    const int zero_len = NN * DP + DP * CP + 2 * NN + 8;

    // d_out layout: s1 [N*CC], norm, e1, sim, total, emb_loss, emb [N*DD]
    float* outS1   = out;
    float* out5    = out + NN * CC;
    float* outEmb  = out + NN * CC + 5;

    const int B = 256;
    auto cdiv = [](int a, int b) { return (a + b - 1) / b; };
    auto gemm_blocks = [&](int M, int Nc) { return cdiv((M / 16) * (Nc / 16), 8); };
    const int vvt_blocks = (NN / 128) * (NN / 64);   // 2048

    // 0) zero pads + accumulators
    k_zero<<<cdiv(zero_len, B), B, 0, stream>>>(XC, zero_len);
    k_pad_wl1<<<cdiv(DD * CC, B), B, 0, stream>>>(Wl1, WL1P);

    // 1) layer 1: Y = adj @ x ; PRE = relu(Y@W1r + x@W1root + b1) ; BN -> XC[:,0:32]
    k_gemm_nn<<<gemm_blocks(NN, FF), B, 0, stream>>>(adj, NN, x, FF, Y, FF, NN, FF, NN);
    k_conv_combine<<<cdiv(NN * HH, B), B, 0, stream>>>(Y, FF, x, FF, W1r, W1root, b1, PRE, HH, HH, FF);
    k_bn_stats<<<HH, B, 0, stream>>>(PRE, HH, MEAN, ISTD);
    k_bn_apply<<<cdiv(NN * HH, B), B, 0, stream>>>(PRE, HH, MEAN, ISTD, g1, be1, XC, 0, HH);

    // 2) layer 2: Y = adj @ x1 ; BN -> XC[:,32:64]
    k_gemm_nn<<<gemm_blocks(NN, HH), B, 0, stream>>>(adj, NN, XC, DP, Y, HH, NN, HH, NN);
    k_conv_combine<<<cdiv(NN * HH, B), B, 0, stream>>>(Y, HH, XC, DP, W2r, W2root, b2, PRE, HH, HH, HH);
    k_bn_stats<<<HH, B, 0, stream>>>(PRE, HH, MEAN, ISTD);
    k_bn_apply<<<cdiv(NN * HH, B), B, 0, stream>>>(PRE, HH, MEAN, ISTD, g2, be2, XC, HH, HH);

    // 3) layer 3: Y = adj @ x2 ; BN -> XC[:,64:269]
    k_gemm_nn<<<gemm_blocks(NN, HH), B, 0, stream>>>(adj, NN, XC + HH, DP, Y, HH, NN, HH, NN);
    k_conv_combine<<<cdiv(NN * CC, B), B, 0, stream>>>(Y, HH, XC + HH, DP, W3r, W3root, b3, PRE, CP, CC, HH);
    k_bn_stats<<<CC, B, 0, stream>>>(PRE, CP, MEAN, ISTD);
    k_bn_apply<<<cdiv(NN * CC, B), B, 0, stream>>>(PRE, CP, MEAN, ISTD, g3, be3, XC, 2 * HH, CC);

    // 4) logits = XC @ WL1P (padded K=272, Nc=208) -> S1P ; softmax etc.
    k_gemm_nn<<<gemm_blocks(NN, CP), B, 0, stream>>>(XC, DP, WL1P, CP, S1P, CP, NN, CP, DP);
    k_softmax_row<<<NN, B, 0, stream>>>(S1P, bl1, outS1, SNP, ACCS + 1);

    // 5) row-normalized x
    k_rownorm_x<<<NN, FF, 0, stream>>>(x, XNo);

    // 6) ||s1 s1^T - adj||_F^2 (fused, LDS/async block tiles)
    k_normloss<<<vvt_blocks, B, 0, stream>>>(S1P, adj, ACCS + 0);

    // 7) similarity: row sums of squares, then masked dot
    k_rowss<<<vvt_blocks, B, 0, stream>>>(XNo, FF, FF, RSX);
    k_rowss<<<vvt_blocks, B, 0, stream>>>(SNP, CP, CP, RSS);
    k_invnorm<<<cdiv(2 * NN, B), B, 0, stream>>>(RSX, RSS);
    k_simdot<<<vvt_blocks, B, 0, stream>>>(XNo, SNP, RSX, RSS, ACCS + 2);

    // 8) team embedding loss + emb output + final scalars
    k_embloss<<<LL, B, 0, stream>>>(XC, teams, reps, ACCS + 3);
    k_write_emb<<<cdiv(NN * DD, B), B, 0, stream>>>(XC, outEmb);
    k_finalize<<<1, 1, 0, stream>>>(ACCS, out5);

    (void)in_sizes; (void)n_in; (void)out_size; (void)ws_size;
}